// ProjectionLayer_90323162235083
// MI455X (gfx1250) — compile-verified
//
#include <hip/hip_runtime.h>
#include <math.h>

typedef __attribute__((ext_vector_type(2))) float v2f;
typedef __attribute__((ext_vector_type(8))) float v8f;

#define TILE_M 128
#define TILE_N 64
#define TILE_K 16
#define LDS_STRIDE 20   // pad: keeps float4 alignment, conflict-free (gcd(20,64) banks pattern distinct over 16 lanes)

// ---------------------------------------------------------------------------
// C[M,N] = epilogue( A[M,K] * B[N,K]^T ), A/B row-major (K contiguous).
// mode 0: plain store. mode 1: tanh(acc + bias[n]).
// Block: 256 threads = 8 waves (wave32). Block tile 128x64.
// Wave grid 4(M) x 2(N); each wave owns a 32x32 region = 2x2 WMMA 16x16 tiles.
// K loop: 16-wide chunks staged in LDS; 4 f32 WMMA k-steps per chunk.
// ---------------------------------------------------------------------------
__global__ __launch_bounds__(256) void pl_gemm_tn_wmma_f32(
    const float* __restrict__ A, const float* __restrict__ B,
    const float* __restrict__ bias, float* __restrict__ C,
    int M, int N, int K, int mode)
{
    __shared__ __align__(16) float As[TILE_M * LDS_STRIDE];
    __shared__ __align__(16) float Bs[TILE_N * LDS_STRIDE];

    const int t    = threadIdx.x;
    const int m0   = blockIdx.y * TILE_M;
    const int n0   = blockIdx.x * TILE_N;
    const int wave = t >> 5;
    const int lane = t & 31;
    const int wm   = (wave >> 1) * 32;   // 0,32,64,96
    const int wn   = (wave & 1) * 32;    // 0,32

    v8f acc00 = {}, acc01 = {}, acc10 = {}, acc11 = {};

    const int ldr  = t >> 2;             // 0..63
    const int ldc  = (t & 3) << 2;       // 0,4,8,12

    for (int kc = 0; kc < K; kc += TILE_K) {
        // Stage A tile (128x16) and B tile (64x16) into LDS, float4-coalesced.
        *(float4*)(&As[ldr * LDS_STRIDE + ldc]) =
            *(const float4*)(A + (size_t)(m0 + ldr) * K + kc + ldc);
        *(float4*)(&As[(ldr + 64) * LDS_STRIDE + ldc]) =
            *(const float4*)(A + (size_t)(m0 + ldr + 64) * K + kc + ldc);
        *(float4*)(&Bs[ldr * LDS_STRIDE + ldc]) =
            *(const float4*)(B + (size_t)(n0 + ldr) * K + kc + ldc);
        __syncthreads();

        // ISA A-frag layout (16x4 f32): lanes 0-15 -> K0/K1, lanes 16-31 -> K2/K3.
        const int fr = lane & 15;
        const int kh = (lane >> 4) << 1; // 0 or 2

        #pragma unroll
        for (int kk = 0; kk < TILE_K; kk += 4) {
            v2f a0, a1, b0, b1;
            a0[0] = As[(wm +      fr) * LDS_STRIDE + kk + kh];
            a0[1] = As[(wm +      fr) * LDS_STRIDE + kk + kh + 1];
            a1[0] = As[(wm + 16 + fr) * LDS_STRIDE + kk + kh];
            a1[1] = As[(wm + 16 + fr) * LDS_STRIDE + kk + kh + 1];
            b0[0] = Bs[(wn +      fr) * LDS_STRIDE + kk + kh];
            b0[1] = Bs[(wn +      fr) * LDS_STRIDE + kk + kh + 1];
            b1[0] = Bs[(wn + 16 + fr) * LDS_STRIDE + kk + kh];
            b1[1] = Bs[(wn + 16 + fr) * LDS_STRIDE + kk + kh + 1];

            acc00 = __builtin_amdgcn_wmma_f32_16x16x4_f32(false, a0, false, b0, (short)0, acc00, false, false);
            acc01 = __builtin_amdgcn_wmma_f32_16x16x4_f32(false, a0, false, b1, (short)0, acc01, false, false);
            acc10 = __builtin_amdgcn_wmma_f32_16x16x4_f32(false, a1, false, b0, (short)0, acc10, false, false);
            acc11 = __builtin_amdgcn_wmma_f32_16x16x4_f32(false, a1, false, b1, (short)0, acc11, false, false);
        }
        __syncthreads();
    }

    // C/D layout: VGPR i -> (M = i + 8*(lane>=16), N = lane&15).
    const int crow = (lane >> 4) << 3;
    const int ccol = lane & 15;

    #pragma unroll
    for (int q = 0; q < 4; ++q) {
        const v8f* accp = (q == 0) ? &acc00 : (q == 1) ? &acc01 : (q == 2) ? &acc10 : &acc11;
        const int rbase = (q < 2) ? wm : wm + 16;
        const int cbase = (q & 1) ? wn + 16 : wn;
        const int col   = n0 + cbase + ccol;
        #pragma unroll
        for (int i = 0; i < 8; ++i) {
            const int row = m0 + rbase + crow + i;
            float v = (*accp)[i];
            if (mode == 1) v = tanhf(v + bias[col]);
            C[(size_t)row * N + col] = v;
        }
    }
}

// ---------------------------------------------------------------------------
// In-place L2 normalize each row of X [rows x D], D == 1024. 1 block/row.
// ---------------------------------------------------------------------------
__global__ __launch_bounds__(256) void pl_l2norm_rows(float* __restrict__ X, int D)
{
    float* row = X + (size_t)blockIdx.x * D;
    const int t = threadIdx.x, wave = t >> 5, lane = t & 31;

    float4 v = *(const float4*)(row + t * 4);
    float ss = v.x * v.x + v.y * v.y + v.z * v.z + v.w * v.w;
    #pragma unroll
    for (int off = 16; off; off >>= 1) ss += __shfl_xor(ss, off, 32);

    __shared__ float wsum[8];
    if (lane == 0) wsum[wave] = ss;
    __syncthreads();

    float tot = 0.f;
    #pragma unroll
    for (int w = 0; w < 8; ++w) tot += wsum[w];
    const float scale = 1.0f / fmaxf(sqrtf(tot), 1e-12f);

    v.x *= scale; v.y *= scale; v.z *= scale; v.w *= scale;
    *(float4*)(row + t * 4) = v;
}

// ---------------------------------------------------------------------------
// Per batch b: argmax_j dot(That[b], V[b,j]) / max(||V[b,j]||, eps).
// One block per batch, one wave per token row (round-robin). Single pass over V.
// Emits idx[b] and 1/||V[b,idx]|| for the gather.
// ---------------------------------------------------------------------------
__global__ __launch_bounds__(256) void pl_argmax_sims(
    const float* __restrict__ V, const float* __restrict__ That,
    int* __restrict__ idx, float* __restrict__ invn, int Ntok, int D)
{
    const int b = blockIdx.x;
    const int t = threadIdx.x, wave = t >> 5, lane = t & 31;

    __shared__ __align__(16) float ts[1024];
    *(float4*)(&ts[t * 4]) = *(const float4*)(That + (size_t)b * D + t * 4);
    __syncthreads();

    float best = -INFINITY, bestinv = 0.f;
    int   bestj = 0x7fffffff;

    for (int j = wave; j < Ntok; j += 8) {
        const float* row = V + ((size_t)b * Ntok + j) * D;
        float dot = 0.f, nrm = 0.f;
        #pragma unroll
        for (int it = 0; it < 8; ++it) {
            const int k = it * 128 + lane * 4;
            float4 vv = *(const float4*)(row + k);
            float4 tv = *(const float4*)(&ts[k]);
            dot += vv.x * tv.x + vv.y * tv.y + vv.z * tv.z + vv.w * tv.w;
            nrm += vv.x * vv.x + vv.y * vv.y + vv.z * vv.z + vv.w * vv.w;
        }
        #pragma unroll
        for (int off = 16; off; off >>= 1) {
            dot += __shfl_xor(dot, off, 32);
            nrm += __shfl_xor(nrm, off, 32);
        }
        const float inv = 1.0f / fmaxf(sqrtf(nrm), 1e-12f);
        const float s   = dot * inv;
        if (s > best) { best = s; bestj = j; bestinv = inv; } // first-max wins within wave
    }

    __shared__ float sbest[8], sinv[8];
    __shared__ int   sj[8];
    if (lane == 0) { sbest[wave] = best; sj[wave] = bestj; sinv[wave] = bestinv; }
    __syncthreads();

    if (t == 0) {
        float bb = sbest[0], bi = sinv[0];
        int bj = sj[0];
        #pragma unroll
        for (int w = 1; w < 8; ++w) {
            if (sbest[w] > bb || (sbest[w] == bb && sj[w] < bj)) {
                bb = sbest[w]; bj = sj[w]; bi = sinv[w];
            }
        }
        idx[b] = bj; invn[b] = bi;
    }
}

// ---------------------------------------------------------------------------
// Vsel[b,:] = V[b, idx[b], :] * invn[b]
// ---------------------------------------------------------------------------
__global__ __launch_bounds__(256) void pl_gather_sel(
    const float* __restrict__ V, const int* __restrict__ idx,
    const float* __restrict__ invn, float* __restrict__ Vsel, int Ntok, int D)
{
    const int b = blockIdx.x;
    const int t = threadIdx.x;
    const int j = idx[b];
    const float s = invn[b];
    const float4* src = (const float4*)(V + ((size_t)b * Ntok + j) * D);
    float4*       dst = (float4*)(Vsel + (size_t)b * D);
    float4 v = src[t];
    v.x *= s; v.y *= s; v.z *= s; v.w *= s;
    dst[t] = v;
}

// ---------------------------------------------------------------------------
extern "C" void kernel_launch(void* const* d_in, const int* in_sizes, int n_in,
                              void* d_out, int out_size, void* d_ws, size_t ws_size,
                              hipStream_t stream)
{
    const float* V    = (const float*)d_in[0];  // [B, N, Dd]
    const float* X    = (const float*)d_in[1];  // [B, Dc]
    const float* W    = (const float*)d_in[2];  // [Dd, Dc]
    const float* bias = (const float*)d_in[3];  // [Dd]
    float* out = (float*)d_out;                 // [B, B]

    const int Dd = in_sizes[3];                 // 1024
    const int Dc = in_sizes[2] / Dd;            // 512
    const int B  = in_sizes[1] / Dc;            // 2048
    const int N  = in_sizes[0] / (B * Dd);      // 196

    // Workspace layout: That [B*Dd] | Vsel [B*Dd] | invn [B] | idx [B]  (~16 MB)
    float* That = (float*)d_ws;
    float* Vsel = That + (size_t)B * Dd;
    float* invn = Vsel + (size_t)B * Dd;
    int*   idx  = (int*)(invn + B);

    // 1) T = tanh(X @ W^T + bias)     (M=B, Nc=Dd, K=Dc)
    {
        dim3 grid(Dd / TILE_N, B / TILE_M);
        pl_gemm_tn_wmma_f32<<<grid, 256, 0, stream>>>(X, W, bias, That, B, Dd, Dc, 1);
    }
    // 2) L2-normalize rows of T (in place)
    pl_l2norm_rows<<<B, 256, 0, stream>>>(That, Dd);
    // 3) streaming argmax over V (the 1.64 GB pass — bandwidth floor of the whole op)
    pl_argmax_sims<<<B, 256, 0, stream>>>(V, That, idx, invn, N, Dd);
    // 4) gather + scale selected rows
    pl_gather_sel<<<B, 256, 0, stream>>>(V, idx, invn, Vsel, N, Dd);
    // 5) out = That @ Vsel^T          (M=B, Nc=B, K=Dd)
    {
        dim3 grid(B / TILE_N, B / TILE_M);
        pl_gemm_tn_wmma_f32<<<grid, 256, 0, stream>>>(That, Vsel, nullptr, out, B, B, Dd, 0);
    }
}